// HyperDiffusion_22393959481939
// MI455X (gfx1250) — compile-verified
//
#include <hip/hip_runtime.h>
#include <hip/hip_bf16.h>

// HyperDiffusion on MI455X (gfx1250):
//  - scatter/gather bound, ~0 matmul FLOPs -> WMMA inapplicable by roofline
//  - all hot buffers (X 51MB, Y 25MB, edge acc 25MB, node acc 51MB) fit in 192MB L2
//  - wave32: 1 wave per incidence entry, 32 lanes x float4 = 128 channels/row
//  - gather rows staged via GLOBAL_LOAD_ASYNC_TO_LDS_B128 with per-wave double
//    buffering (ASYNCcnt pipelining); degrees prefetched via global_prefetch_b8
//  - unsafeAtomicAdd -> native global_atomic_add_f32 (L2-resident rows)

#define CH    128
#define CH4   (CH / 4)
#define ROWB  (CH * 4)          // 512 bytes per feature row
#define WPB   8                 // waves per 256-thread block

__device__ __forceinline__ void fadd(float* p, float v) {
    unsafeAtomicAdd(p, v);      // native global_atomic_add_f32, no CAS loop
}

__global__ void k_zero_f4(float4* __restrict__ p, long n4) {
    long i = (long)blockIdx.x * blockDim.x + threadIdx.x;
    long s = (long)gridDim.x * blockDim.x;
    const float4 z = make_float4(0.f, 0.f, 0.f, 0.f);
    for (; i < n4; i += s) p[i] = z;
}

__global__ void k_zero_u32(unsigned* __restrict__ p, long n) {
    long i = (long)blockIdx.x * blockDim.x + threadIdx.x;
    long s = (long)gridDim.x * blockDim.x;
    for (; i < n; i += s) p[i] = 0u;
}

// degree histograms: deg_v[node_idx[i]]++, deg_e[edge_idx[i]]++
__global__ void k_deg(const int* __restrict__ nidx, const int* __restrict__ eidx,
                      unsigned* __restrict__ degv, unsigned* __restrict__ dege, int nnz) {
    int i = blockIdx.x * blockDim.x + threadIdx.x;
    int s = gridDim.x * blockDim.x;
    for (; i < nnz; i += s) {
        atomicAdd(&degv[nidx[i]], 1u);
        atomicAdd(&dege[eidx[i]], 1u);
    }
}

// softmax over axis 0 of lazy_weights [2, CH] -> sw[0..CH)=sw0, sw[CH..2CH)=sw1
__global__ void k_softmax(const float* __restrict__ lw, float* __restrict__ sw) {
    int c = threadIdx.x;             // launched with CH threads
    float w0 = lw[c], w1 = lw[CH + c];
    float m  = fmaxf(w0, w1);
    float e0 = __expf(w0 - m), e1 = __expf(w1 - m);
    float inv = 1.0f / (e0 + e1);
    sw[c]      = e0 * inv;
    sw[CH + c] = e1 * inv;
}

// Segment-sum phase:  dst[didx[i]] += src[sidx[i]] * invdeg(sdeg[sidx[i]])
// One wave32 per incidence entry; the 512B source row for iteration i+1 is
// streamed into LDS by the async unit (GLOBAL_LOAD_ASYNC_TO_LDS_B128) while
// iteration i's 128 L2-resident f32 atomics drain. Per-wave double buffer,
// ordered with s_wait_asynccnt (async loads complete in order).
__global__ void __launch_bounds__(WPB * 32)
k_scatter(const float* __restrict__ src, const int* __restrict__ sidx,
          const int* __restrict__ didx, const unsigned* __restrict__ sdeg,
          float* __restrict__ dst, int nnz) {
    __shared__ float lbuf[WPB][2][CH];            // 2 x 512B staging per wave

    const int  lane = threadIdx.x & 31;
    const int  wib  = threadIdx.x >> 5;
    const long wave = ((long)blockIdx.x * blockDim.x + threadIdx.x) >> 5;
    const long nw   = ((long)gridDim.x * blockDim.x) >> 5;

    // per-lane LDS byte addresses of the two staging slots (AS3 offset = low 32b)
    const unsigned lds0 = (unsigned)(uintptr_t)(&lbuf[wib][0][lane << 2]);
    const unsigned lds1 = (unsigned)(uintptr_t)(&lbuf[wib][1][lane << 2]);

    long i = wave;
    if (i >= nnz) return;

    // prologue: stage first row into buffer 0
    {
        const unsigned s0   = (unsigned)sidx[i];
        const unsigned moff = s0 * (unsigned)ROWB + ((unsigned)lane << 4);
        asm volatile("global_load_async_to_lds_b128 %0, %1, %2"
                     :: "v"(lds0), "v"(moff), "s"(src) : "memory");
    }

    int par = 0;
    for (; i < nnz; i += nw) {
        const int s = sidx[i];
        const int d = didx[i];

        const long ip = i + nw;
        if (ip < nnz) {
            const int sp = sidx[ip];
            __builtin_prefetch(&sdeg[sp], 0, 3);               // global_prefetch_b8
            const unsigned moff = (unsigned)sp * (unsigned)ROWB + ((unsigned)lane << 4);
            const unsigned ldsN = par ? lds0 : lds1;           // fill the other slot
            asm volatile("global_load_async_to_lds_b128 %0, %1, %2"
                         :: "v"(ldsN), "v"(moff), "s"(src) : "memory");
            // 2 outstanding; oldest (current row) done when count <= 1
            asm volatile("s_wait_asynccnt 0x1" ::: "memory");
        } else {
            asm volatile("s_wait_asynccnt 0x0" ::: "memory");  // drain last transfer
        }

        const unsigned dg = sdeg[s];
        const float inv = dg ? 1.0f / (float)dg : 0.0f;

        const float4 x = *(const float4*)(&lbuf[wib][par][lane << 2]);  // ds_load_b128
        float* o = dst + (long)d * CH + (lane << 2);
        fadd(o + 0, x.x * inv);
        fadd(o + 1, x.y * inv);
        fadd(o + 2, x.z * inv);
        fadd(o + 3, x.w * inv);

        par ^= 1;
    }
}

// In-place lazy mix:  acc = sw0[c]*acc + sw1[c]*other   (rows of CH channels)
__global__ void k_mix(float4* __restrict__ acc, const float4* __restrict__ other,
                      const float* __restrict__ sw, long n4) {
    long i = (long)blockIdx.x * blockDim.x + threadIdx.x;
    long s = (long)gridDim.x * blockDim.x;
    const float4* s0v = (const float4*)sw;
    const float4* s1v = (const float4*)(sw + CH);
    for (; i < n4; i += s) {
        const int c4 = (int)(i & (CH4 - 1));     // CH4 == 32, power of two
        const float4 s0 = s0v[c4];
        const float4 s1 = s1v[c4];
        float4 a = acc[i];
        const float4 o = other[i];
        a.x = fmaf(s0.x, a.x, s1.x * o.x);
        a.y = fmaf(s0.y, a.y, s1.y * o.y);
        a.z = fmaf(s0.z, a.z, s1.z * o.z);
        a.w = fmaf(s0.w, a.w, s1.w * o.w);
        acc[i] = a;
    }
}

extern "C" void kernel_launch(void* const* d_in, const int* in_sizes, int n_in,
                              void* d_out, int out_size, void* d_ws, size_t ws_size,
                              hipStream_t stream) {
    const float* X   = (const float*)d_in[0];   // [N, CH]
    const float* Y   = (const float*)d_in[1];   // [E, CH]
    const float* lw  = (const float*)d_in[2];   // [2, CH]
    const int*  nidx = (const int*)d_in[3];     // [NNZ]
    const int*  eidx = (const int*)d_in[4];     // [NNZ]

    const int N   = in_sizes[0] / CH;
    const int E   = in_sizes[1] / CH;
    const int NNZ = in_sizes[3];

    float* nodeOut = (float*)d_out;                  // output 0: node_feat [N, CH]
    float* edgeOut = (float*)d_out + (long)N * CH;   // output 1: edge_feat [E, CH]

    // workspace layout: deg_v u32[N] | deg_e u32[E] | sw f32[2*CH] (16B aligned)
    unsigned* degv = (unsigned*)d_ws;
    unsigned* dege = degv + N;
    size_t swOff = (((size_t)(N + E) * sizeof(unsigned)) + 15) & ~(size_t)15;
    float* sw = (float*)((char*)d_ws + swOff);

    // 1) zero output accumulators + degree counters
    k_zero_f4 <<<4096, 256, 0, stream>>>((float4*)d_out, (long)out_size / 4);
    k_zero_u32<<< 512, 256, 0, stream>>>(degv, (long)(N + E));

    // 2) degrees + softmax weights
    k_deg    <<<2048, 256, 0, stream>>>(nidx, eidx, degv, dege, NNZ);
    k_softmax<<<   1,  CH, 0, stream>>>(lw, sw);

    // 3) v2e: edge_acc[e] += X[v] * D_v^-1[v]
    k_scatter<<<2048, 256, 0, stream>>>(X, nidx, eidx, degv, edgeOut, NNZ);

    // 4) edge_feat = sw0*edge_acc + sw1*Y   (this IS the edge output)
    k_mix<<<4096, 256, 0, stream>>>((float4*)edgeOut, (const float4*)Y, sw,
                                    (long)E * CH4);

    // 5) e2v: node_acc[v] += edge_feat[e] * D_e^-1[e]
    k_scatter<<<2048, 256, 0, stream>>>(edgeOut, eidx, nidx, dege, nodeOut, NNZ);

    // 6) node_feat = sw0*node_acc + sw1*X
    k_mix<<<4096, 256, 0, stream>>>((float4*)nodeOut, (const float4*)X, sw,
                                    (long)N * CH4);
}